// newGCN_85100482003075
// MI455X (gfx1250) — compile-verified
//
#include <hip/hip_runtime.h>
#include <hip/hip_bf16.h>

#define HDIM 96
#define NGRAPH 500
#define OUTD 10
#define BN_EPS 1e-5f

typedef __attribute__((ext_vector_type(2))) float v2f;
typedef __attribute__((ext_vector_type(8))) float v8f;

// ---------------------------------------------------------------------------
// WMMA GEMM:  Y[M x 96] = X[M x 96] @ W[96 x 96] (+ bias) (+ relu)
// One wave per 32-row tile (2 x 16-row WMMA tiles); 6 column tiles of 16.
// W staged in LDS pair-interleaved so every B fragment is one ds_load_b64.
// A fragments are unconditional global_load_b64 (clamped row + 0/1 mask).
// Uses V_WMMA_F32_16X16X4_F32 (fp32 matrix path, matches fp32 reference).
// ---------------------------------------------------------------------------
__global__ __launch_bounds__(256) void gemm_wmma_f32(
    const float* __restrict__ X, const float* __restrict__ W,
    const float* __restrict__ bias, float* __restrict__ Y, int M, int relu) {
  // Pair-interleaved: Ws[(k>>1)*192 + c*2 + (k&1)] = W[k][c]
  __shared__ float Ws[(HDIM / 2) * (HDIM * 2)];  // 36 KB
  for (int i = threadIdx.x; i < HDIM * HDIM; i += 256) {
    int k = i / HDIM, c = i - k * HDIM;
    Ws[(k >> 1) * (2 * HDIM) + c * 2 + (k & 1)] = W[i];
  }
  __syncthreads();

  const int wave = threadIdx.x >> 5;
  const int lane = threadIdx.x & 31;
  const long r0 = ((long)blockIdx.x * 8 + wave) * 32;  // two 16-row tiles
  if (r0 >= M) return;  // uniform per wave -> EXEC all-1 for WMMA

  const int m = lane & 15;
  const int kh = (lane >> 4) * 2;  // lanes 0-15: K,K+1 ; lanes 16-31: K+2,K+3

  const long rowA = r0 + m;
  const long rowB = r0 + 16 + m;
  const float maskA = (rowA < M) ? 1.0f : 0.0f;
  const float maskB = (rowB < M) ? 1.0f : 0.0f;
  const float* __restrict__ xa = X + (rowA < M ? rowA : (long)(M - 1)) * HDIM;
  const float* __restrict__ xb = X + (rowB < M ? rowB : (long)(M - 1)) * HDIM;

  v8f acc[12];
#pragma unroll
  for (int t = 0; t < 12; ++t) acc[t] = (v8f){0.f, 0.f, 0.f, 0.f, 0.f, 0.f, 0.f, 0.f};

  for (int k = 0; k < HDIM; k += 4) {
    v2f a0 = *(const v2f*)(xa + k + kh);  // unconditional b64 load
    v2f a1 = *(const v2f*)(xb + k + kh);
    a0 *= maskA;
    a1 *= maskB;
    const float* __restrict__ wrow = &Ws[((k + kh) >> 1) * (2 * HDIM)];
#pragma unroll
    for (int t = 0; t < 6; ++t) {
      v2f b = *(const v2f*)(wrow + (t * 16 + m) * 2);  // one ds_load_b64
      acc[t] = __builtin_amdgcn_wmma_f32_16x16x4_f32(
          false, a0, false, b, (short)0, acc[t], false, false);
      acc[6 + t] = __builtin_amdgcn_wmma_f32_16x16x4_f32(
          false, a1, false, b, (short)0, acc[6 + t], false, false);
    }
  }

  const int roff = (lane >> 4) ? 8 : 0;
#pragma unroll
  for (int half = 0; half < 2; ++half) {
    const long rbase = r0 + half * 16 + roff;
#pragma unroll
    for (int t = 0; t < 6; ++t) {
      const int col = t * 16 + m;
      const float bv = bias ? bias[col] : 0.0f;
#pragma unroll
      for (int r = 0; r < 8; ++r) {
        const long rr = rbase + r;
        if (rr < M) {
          float v = acc[half * 6 + t][r] + bv;
          if (relu) v = fmaxf(v, 0.0f);
          Y[rr * HDIM + col] = v;
        }
      }
    }
  }
}

// ---------------------------------------------------------------------------
// Small helpers
// ---------------------------------------------------------------------------
__global__ void fill_f32(float* p, float v, long n) {
  long i = (long)blockIdx.x * blockDim.x + threadIdx.x;
  long stride = (long)gridDim.x * blockDim.x;
  for (; i < n; i += stride) p[i] = v;
}

__global__ void deg_count(const int* __restrict__ dst, float* deg, int E) {
  int e = blockIdx.x * blockDim.x + threadIdx.x;
  if (e < E) atomicAdd(&deg[dst[e]], 1.0f);
}

__global__ void deg_finalize(float* d, int N) {
  int i = blockIdx.x * blockDim.x + threadIdx.x;
  if (i < N) d[i] = rsqrtf(d[i]);  // dinv = deg^-1/2
}

// out[i][:] = (dinv[i]^2) * h[i][:] + bias[:]
__global__ void init_agg(const float* __restrict__ h, const float* __restrict__ dinv,
                         const float* __restrict__ bias, float* __restrict__ out, long N) {
  long idx = (long)blockIdx.x * blockDim.x + threadIdx.x;
  long total = N * HDIM;
  long stride = (long)gridDim.x * blockDim.x;
  for (; idx < total; idx += stride) {
    long i = idx / HDIM;
    int c = (int)(idx - i * HDIM);
    float dw = dinv[i];
    out[idx] = dw * dw * h[idx] + bias[c];
  }
}

// One wave per edge: out[dst] += dinv[src]*dinv[dst]*h[src]
__global__ __launch_bounds__(128) void edge_scatter(
    const float* __restrict__ h, const int* __restrict__ src,
    const int* __restrict__ dst, const float* __restrict__ dinv,
    float* __restrict__ out, int E) {
  int e = blockIdx.x * 4 + (threadIdx.x >> 5);
  if (e >= E) return;
  int lane = threadIdx.x & 31;
  int s = src[e], d = dst[e];
  float w = dinv[s] * dinv[d];
  const float* __restrict__ hs = h + (long)s * HDIM;
  float* od = out + (long)d * HDIM;
#pragma unroll
  for (int c = lane; c < HDIM; c += 32) atomicAdd(&od[c], w * hs[c]);
}

// Per-column mean / rsqrt(var+eps). One block per column.
__global__ __launch_bounds__(256) void bn_stat(const float* __restrict__ X, int M,
                                               float* mu, float* rs) {
  __shared__ float ssum[256];
  __shared__ float ssq[256];
  int c = blockIdx.x;
  float s = 0.f, q = 0.f;
  for (int i = threadIdx.x; i < M; i += 256) {
    float v = X[(long)i * HDIM + c];
    s += v;
    q += v * v;
  }
  ssum[threadIdx.x] = s;
  ssq[threadIdx.x] = q;
  __syncthreads();
  for (int off = 128; off > 0; off >>= 1) {
    if (threadIdx.x < off) {
      ssum[threadIdx.x] += ssum[threadIdx.x + off];
      ssq[threadIdx.x] += ssq[threadIdx.x + off];
    }
    __syncthreads();
  }
  if (threadIdx.x == 0) {
    float mean = ssum[0] / (float)M;
    float var = ssq[0] / (float)M - mean * mean;
    mu[c] = mean;
    rs[c] = rsqrtf(fmaxf(var, 0.0f) + BN_EPS);
  }
}

// In-place x = relu((x - mu) * rs * g + b)
__global__ void bn_apply_relu(float* X, long M, const float* __restrict__ mu,
                              const float* __restrict__ rs, const float* __restrict__ g,
                              const float* __restrict__ b) {
  long idx = (long)blockIdx.x * blockDim.x + threadIdx.x;
  long total = M * HDIM;
  long stride = (long)gridDim.x * blockDim.x;
  for (; idx < total; idx += stride) {
    int c = (int)(idx % HDIM);
    float v = (X[idx] - mu[c]) * rs[c] * g[c] + b[c];
    X[idx] = fmaxf(v, 0.0f);
  }
}

// Pooling: one wave per node accumulates into pooled[batch[i]]; lane0 bumps cnt.
__global__ __launch_bounds__(128) void pool_acc(const float* __restrict__ X,
                                                const int* __restrict__ batch,
                                                float* pooled, float* cnt, int N) {
  int i = blockIdx.x * 4 + (threadIdx.x >> 5);
  if (i >= N) return;
  int lane = threadIdx.x & 31;
  int g = batch[i];
  const float* __restrict__ xi = X + (long)i * HDIM;
  float* pg = pooled + (long)g * HDIM;
#pragma unroll
  for (int c = lane; c < HDIM; c += 32) atomicAdd(&pg[c], xi[c]);
  if (lane == 0) atomicAdd(&cnt[g], 1.0f);
}

__global__ void pool_div(float* pooled, const float* __restrict__ cnt) {
  int idx = blockIdx.x * blockDim.x + threadIdx.x;
  if (idx < NGRAPH * HDIM) pooled[idx] /= fmaxf(cnt[idx / HDIM], 1.0f);
}

// Final head: out[M x 10] = X[M x 96] @ W3[96 x 10] + b3
__global__ void head_out(const float* __restrict__ X, const float* __restrict__ W3,
                         const float* __restrict__ b3, float* __restrict__ out, int M) {
  int idx = blockIdx.x * blockDim.x + threadIdx.x;
  if (idx >= M * OUTD) return;
  int i = idx / OUTD, j = idx % OUTD;
  float s = b3[j];
  const float* xi = X + (long)i * HDIM;
#pragma unroll 4
  for (int k = 0; k < HDIM; ++k) s += xi[k] * W3[k * OUTD + j];
  out[idx] = s;
}

// ---------------------------------------------------------------------------
// Launch sequence
// ---------------------------------------------------------------------------
extern "C" void kernel_launch(void* const* d_in, const int* in_sizes, int n_in,
                              void* d_out, int out_size, void* d_ws, size_t ws_size,
                              hipStream_t stream) {
  const float* x_in  = (const float*)d_in[0];
  const int*   eidx  = (const int*)d_in[1];
  const int*   batch = (const int*)d_in[2];
  const float* convW = (const float*)d_in[3];
  const float* convB = (const float*)d_in[4];
  const float* bnG   = (const float*)d_in[5];
  const float* bnB   = (const float*)d_in[6];
  const float* W1    = (const float*)d_in[7];
  const float* b1    = (const float*)d_in[8];
  const float* g1    = (const float*)d_in[9];
  const float* be1   = (const float*)d_in[10];
  const float* W2    = (const float*)d_in[11];
  const float* b2    = (const float*)d_in[12];
  const float* W3    = (const float*)d_in[13];
  const float* b3    = (const float*)d_in[14];
  float* out = (float*)d_out;

  const int N = in_sizes[0] / HDIM;
  const int E = in_sizes[1] / 2;
  const int* src = eidx;
  const int* dst = eidx + E;

  // Workspace carve-up (floats)
  float* f0     = (float*)d_ws;            // N x 96  (aggregated / current features)
  float* f1     = f0 + (long)N * HDIM;     // N x 96  (h = x @ W)
  float* dinv   = f1 + (long)N * HDIM;     // N
  float* mu     = dinv + N;                // 96
  float* rs     = mu + HDIM;               // 96
  float* pooled = rs + HDIM;               // 500 x 96
  float* cnt    = pooled + NGRAPH * HDIM;  // 500
  float* hb     = cnt + NGRAPH;            // 500 x 96 (head buffer)

  const long NH = (long)N * HDIM;
  const int gelem = (int)((NH + 255) / 256);

  // Degrees with self-loop, then dinv = rsqrt(deg)
  fill_f32<<<(N + 255) / 256, 256, 0, stream>>>(dinv, 1.0f, N);
  deg_count<<<(E + 255) / 256, 256, 0, stream>>>(dst, dinv, E);
  deg_finalize<<<(N + 255) / 256, 256, 0, stream>>>(dinv, N);

  const int gemm_blocks_N = (int)(((long)(N + 31) / 32 + 7) / 8);
  const float* cur = x_in;
  for (int layer = 0; layer < 6; ++layer) {
    const float* Wl = convW + (long)layer * HDIM * HDIM;
    const float* Bl = convB + (long)layer * HDIM;
    // h = cur @ W  (WMMA, no bias/relu)
    gemm_wmma_f32<<<gemm_blocks_N, 256, 0, stream>>>(cur, Wl, nullptr, f1, N, 0);
    // agg = self_w * h + bias
    init_agg<<<gelem, 256, 0, stream>>>(f1, dinv, Bl, f0, N);
    // agg[dst] += ew * h[src]
    edge_scatter<<<(E + 3) / 4, 128, 0, stream>>>(f1, src, dst, dinv, f0, E);
    if (layer < 5) {
      bn_stat<<<HDIM, 256, 0, stream>>>(f0, N, mu, rs);
      bn_apply_relu<<<gelem, 256, 0, stream>>>(f0, N, mu, rs, bnG + layer * HDIM,
                                               bnB + layer * HDIM);
    }
    cur = f0;
  }

  // Global mean pool
  fill_f32<<<(NGRAPH * HDIM + NGRAPH + 255) / 256, 256, 0, stream>>>(pooled, 0.0f,
                                                                     NGRAPH * HDIM + NGRAPH);
  pool_acc<<<(N + 3) / 4, 128, 0, stream>>>(f0, batch, pooled, cnt, N);
  pool_div<<<(NGRAPH * HDIM + 255) / 256, 256, 0, stream>>>(pooled, cnt);

  // Head: Linear -> BN -> ReLU -> Linear -> ReLU -> Linear
  const int gemm_blocks_P = ((NGRAPH + 31) / 32 + 7) / 8;
  gemm_wmma_f32<<<gemm_blocks_P, 256, 0, stream>>>(pooled, W1, b1, hb, NGRAPH, 0);
  bn_stat<<<HDIM, 256, 0, stream>>>(hb, NGRAPH, mu, rs);
  bn_apply_relu<<<(NGRAPH * HDIM + 255) / 256, 256, 0, stream>>>(hb, NGRAPH, mu, rs, g1, be1);
  gemm_wmma_f32<<<gemm_blocks_P, 256, 0, stream>>>(hb, W2, b2, pooled, NGRAPH, 1);
  head_out<<<(NGRAPH * OUTD + 255) / 256, 256, 0, stream>>>(pooled, W3, b3, out, NGRAPH);
}